// DeltaE2000Loss_15728170238214
// MI455X (gfx1250) — compile-verified
//
#include <hip/hip_runtime.h>
#include <math.h>

typedef float v2f __attribute__((ext_vector_type(2)));
typedef float v8f __attribute__((ext_vector_type(8)));

#define TWO_PI_F   6.2831853071795864769f
#define PI_F       3.1415926535897932385f
#define HALF_PI_F  1.5707963267948966192f
#define POW25_7_F  6103515625.0f   // 25^7

// ---------------------------------------------------------------------------
// Raw hardware TRANS-pipe helpers (single v_* instructions, no libm slow path)
// ---------------------------------------------------------------------------
__device__ __forceinline__ float h_rcp (float x) { return __builtin_amdgcn_rcpf(x);  }
__device__ __forceinline__ float h_rsq (float x) { return __builtin_amdgcn_rsqf(x);  }
__device__ __forceinline__ float h_sqrt(float x) { return __builtin_amdgcn_sqrtf(x); }
__device__ __forceinline__ float h_exp2(float x) { return __builtin_amdgcn_exp2f(x); }  // v_exp_f32
__device__ __forceinline__ float h_log2(float x) { return __builtin_amdgcn_logf(x);  }  // v_log_f32 (log2)
// x > 0 required:
__device__ __forceinline__ float h_pow (float x, float y) { return h_exp2(y * h_log2(x)); }
__device__ __forceinline__ float h_cbrt(float x) { return h_exp2(0.333333333333f * h_log2(x)); }

// Branchless fast atan2 (polynomial, ~1e-6 rad abs error), matches
// atan2f quadrant conventions; returns 0 at (0,0) like jnp.arctan2.
__device__ __forceinline__ float fast_atan2(float y, float x) {
    float ax = fabsf(x), ay = fabsf(y);
    float mx = fmaxf(ax, ay), mn = fminf(ax, ay);
    float r  = (mx > 0.0f) ? mn * h_rcp(mx) : 0.0f;      // [0,1]
    float r2 = r * r;
    float p  = -0.0117212f;
    p = fmaf(p, r2,  0.05265332f);
    p = fmaf(p, r2, -0.11643287f);
    p = fmaf(p, r2,  0.19354346f);
    p = fmaf(p, r2, -0.33262347f);
    p = fmaf(p, r2,  0.99997726f);
    float a = r * p;                                     // atan(mn/mx)
    a = (ay > ax) ? (HALF_PI_F - a) : a;
    a = (x < 0.0f) ? (PI_F - a) : a;
    return (y < 0.0f) ? -a : a;
}

// ---------------------------------------------------------------------------
// sRGB -> Lab -> CIEDE2000, replicating the JAX reference (incl. its
// "theta in degrees passed to sin" quirk).
// ---------------------------------------------------------------------------
__device__ __forceinline__ float srgb_lin(float c) {
    return (c > 0.04045f) ? h_pow((c + 0.055f) * (1.0f / 1.055f), 2.4f)
                          : c * (1.0f / 12.92f);
}

__device__ __forceinline__ float lab_f(float t) {
    return (t > 0.008856f) ? h_cbrt(t) : fmaf(7.787f, t, 16.0f / 116.0f);
}

__device__ __forceinline__ void rgb_to_lab(float r, float g, float b,
                                           float& L, float& A, float& Bb) {
    float lr = srgb_lin(r), lg = srgb_lin(g), lb = srgb_lin(b);
    float X = 0.4124564f * lr + 0.3575761f * lg + 0.1804375f * lb;
    float Y = 0.2126729f * lr + 0.7151522f * lg + 0.0721750f * lb;
    float Z = 0.0193339f * lr + 0.1191920f * lg + 0.9503041f * lb;
    X *= (1.0f / 0.95047f);
    Z *= (1.0f / 1.08883f);
    float fx = lab_f(X), fy = lab_f(Y), fz = lab_f(Z);
    L  = fmaf(116.0f, fy, -16.0f);
    A  = 500.0f * (fx - fy);
    Bb = 200.0f * (fy - fz);
}

__device__ __forceinline__ float pow7(float x) {
    float x2 = x * x;
    float x4 = x2 * x2;
    return x4 * x2 * x;
}

__device__ __forceinline__ float hue_mod(float b, float a) {
    float h = fast_atan2(b, a);             // [-pi, pi]
    return (h < 0.0f) ? h + TWO_PI_F : h;   // == h % 2pi
}

__device__ __forceinline__ float delta_e_2000(float L1, float a1, float b1,
                                              float L2, float a2, float b2) {
    float C1 = h_sqrt(fmaf(a1, a1, b1 * b1));
    float C2 = h_sqrt(fmaf(a2, a2, b2 * b2));
    float C_avg  = 0.5f * (C1 + C2);
    float C_avg7 = pow7(C_avg);
    float G = 0.5f * (1.0f - h_sqrt(C_avg7 * h_rcp(C_avg7 + POW25_7_F)));
    float a1p = a1 * (1.0f + G);
    float a2p = a2 * (1.0f + G);
    float C1p = h_sqrt(fmaf(a1p, a1p, b1 * b1));
    float C2p = h_sqrt(fmaf(a2p, a2p, b2 * b2));
    float h1p = hue_mod(b1, a1p);
    float h2p = hue_mod(b2, a2p);

    float dL = L2 - L1;
    float dC = C2p - C1p;
    float dh = h2p - h1p;
    if (fabsf(dh) > PI_F)
        dh = (h2p <= h1p) ? dh + TWO_PI_F : dh - TWO_PI_F;
    float dH = 2.0f * h_sqrt(C1p * C2p) * __sinf(0.5f * dh);

    float L_avg  = 0.5f * (L1 + L2);
    float Cp_avg = 0.5f * (C1p + C2p);
    float h_avg  = (fabsf(h1p - h2p) > PI_F) ? 0.5f * (h1p + h2p + TWO_PI_F)
                                             : 0.5f * (h1p + h2p);

    float T = 1.0f
            - 0.17f * __cosf(h_avg - PI_F / 6.0f)
            + 0.24f * __cosf(2.0f * h_avg)
            + 0.32f * __cosf(fmaf(3.0f, h_avg, PI_F / 30.0f))
            - 0.20f * __cosf(fmaf(4.0f, h_avg, -7.0f * PI_F / 20.0f));

    float hdeg  = h_avg * (180.0f / PI_F);
    float targ  = (hdeg - 275.0f) * (1.0f / 25.0f);
    float theta = 30.0f * __expf(-(targ * targ));

    float Cp_avg7 = pow7(Cp_avg);
    float RC = 2.0f * h_sqrt(Cp_avg7 * h_rcp(Cp_avg7 + POW25_7_F));
    float RT = -__sinf(2.0f * theta) * RC;   // matches reference quirk

    float Lm50  = L_avg - 50.0f;
    float Lm50s = Lm50 * Lm50;
    float SL = fmaf(0.015f * Lm50s, h_rsq(20.0f + Lm50s), 1.0f);
    float SC = fmaf(0.045f, Cp_avg, 1.0f);
    float SH = fmaf(0.015f * Cp_avg, T, 1.0f);

    float tL = dL * h_rcp(SL), tC = dC * h_rcp(SC), tH = dH * h_rcp(SH);
    return h_sqrt(fmaf(tL, tL, fmaf(tC, tC, fmaf(tH, tH, RT * tC * tH))));
}

__device__ __forceinline__ float combined_loss(float pr, float pg, float pb,
                                               float tr, float tg, float tb) {
    float dr = pr - tr, dg = pg - tg, db = pb - tb;
    float mse = (fmaf(dr, dr, fmaf(dg, dg, db * db))) * (1.0f / 3.0f);
    float L1, A1, B1, L2, A2, B2;
    rgb_to_lab(pr, pg, pb, L1, A1, B1);
    rgb_to_lab(tr, tg, tb, L2, A2, B2);
    float de = delta_e_2000(L1, A1, B1, L2, A2, B2);
    return fmaf(0.2f, mse, 0.8f * de);
}

// ---------------------------------------------------------------------------
// Wave reduction: V_WMMA_F32_16X16X4_F32 (A = ones, B holds 32 lane accs,
// C = 0) folds lanes n and n+16 on the matrix pipe; 4 XOR ds_swizzle steps
// (group-of-32 mode: offset = (xor<<10)|0x1F) finish 16 -> 1.
// EXEC must be all ones: every thread calls this unconditionally.
// ---------------------------------------------------------------------------
template <int XMASK>
__device__ __forceinline__ float xor_add(float s) {
    int o = __builtin_amdgcn_ds_swizzle(__float_as_int(s), (XMASK << 10) | 0x1F);
    return s + __int_as_float(o);
}

__device__ __forceinline__ float wave_reduce_sum(float acc) {
    v2f a; a.x = 1.0f; a.y = 1.0f;   // A = ones(16x4)
    v2f b; b.x = acc;  b.y = 0.0f;   // 32 accumulators live in b0
    v8f c = {};
    c = __builtin_amdgcn_wmma_f32_16x16x4_f32(
            false, a, false, b, (short)0, c, false, false);
    float s = c[0];                  // lane i: acc[i%16] + acc[i%16+16]
    s = xor_add<8>(s);
    s = xor_add<4>(s);
    s = xor_add<2>(s);
    s = xor_add<1>(s);
    return s;                        // full 32-lane sum in every lane
}

// ---------------------------------------------------------------------------
// Kernel 1: grid-stride over pixel quads (3 x float4 = 4 pixels / tensor),
// fully coalesced global_load_b128; block partial -> d_ws[block].
// ---------------------------------------------------------------------------
__global__ __launch_bounds__(256)
void de2000_partial_kernel(const float* __restrict__ pred,
                           const float* __restrict__ target,
                           float* __restrict__ partials,
                           long long B) {
    const long long nquad   = B >> 2;
    const long long gtid    = (long long)blockIdx.x * blockDim.x + threadIdx.x;
    const long long gstride = (long long)gridDim.x * blockDim.x;

    float acc = 0.0f;

    for (long long q = gtid; q < nquad; q += gstride) {
        const float4* p4 = reinterpret_cast<const float4*>(pred   + q * 12);
        const float4* t4 = reinterpret_cast<const float4*>(target + q * 12);
        float4 p0 = p4[0], p1 = p4[1], p2 = p4[2];
        float4 t0 = t4[0], t1 = t4[1], t2 = t4[2];
        acc += combined_loss(p0.x, p0.y, p0.z, t0.x, t0.y, t0.z);
        acc += combined_loss(p0.w, p1.x, p1.y, t0.w, t1.x, t1.y);
        acc += combined_loss(p1.z, p1.w, p2.x, t1.z, t1.w, t2.x);
        acc += combined_loss(p2.y, p2.z, p2.w, t2.y, t2.z, t2.w);
    }

    // Tail pixels (B not divisible by 4) — scalar path.
    for (long long i = (nquad << 2) + gtid; i < B; i += gstride) {
        const float* p = pred   + i * 3;
        const float* t = target + i * 3;
        acc += combined_loss(p[0], p[1], p[2], t[0], t[1], t[2]);
    }

    float wsum = wave_reduce_sum(acc);

    __shared__ float lds[8];
    const int lane = threadIdx.x & 31;
    const int wave = threadIdx.x >> 5;
    if (lane == 0) lds[wave] = wsum;
    __syncthreads();
    if (threadIdx.x == 0) {
        float bsum = 0.0f;
        #pragma unroll
        for (int w = 0; w < 8; ++w) bsum += lds[w];
        partials[blockIdx.x] = bsum;
    }
}

// ---------------------------------------------------------------------------
// Kernel 2: one block deterministically folds block partials, scales by 1/B.
// ---------------------------------------------------------------------------
__global__ __launch_bounds__(256)
void de2000_final_kernel(const float* __restrict__ partials, int nparts,
                         float* __restrict__ out, float inv_B) {
    float acc = 0.0f;
    for (int i = threadIdx.x; i < nparts; i += 256)
        acc += partials[i];

    float wsum = wave_reduce_sum(acc);

    __shared__ float lds[8];
    const int lane = threadIdx.x & 31;
    const int wave = threadIdx.x >> 5;
    if (lane == 0) lds[wave] = wsum;
    __syncthreads();
    if (threadIdx.x == 0) {
        float total = 0.0f;
        #pragma unroll
        for (int w = 0; w < 8; ++w) total += lds[w];
        out[0] = total * inv_B;
    }
}

// ---------------------------------------------------------------------------
extern "C" void kernel_launch(void* const* d_in, const int* in_sizes, int n_in,
                              void* d_out, int out_size, void* d_ws, size_t ws_size,
                              hipStream_t stream) {
    const float* pred   = (const float*)d_in[0];
    const float* target = (const float*)d_in[1];
    float* out      = (float*)d_out;
    float* partials = (float*)d_ws;

    const long long B = (long long)in_sizes[0] / 3;   // 16,777,216

    const int block   = 256;
    const int nblocks = 4096;   // 1M threads; 4 quad-iterations each

    de2000_partial_kernel<<<nblocks, block, 0, stream>>>(pred, target, partials, B);
    de2000_final_kernel<<<1, block, 0, stream>>>(partials, nblocks, out,
                                                 1.0f / (float)B);
}